// Layer_37606733644003
// MI455X (gfx1250) — compile-verified
//
#include <hip/hip_runtime.h>

typedef float v2f __attribute__((ext_vector_type(2)));
typedef float v8f __attribute__((ext_vector_type(8)));

// out[n][o] = relu( sum_i x[n][i] * W[o][i] )
// One wave -> 16-row x 32-col output tile via two 16x16 f32 WMMA accumulators,
// K=32 as 8 chunks of V_WMMA_F32_16X16X4_F32.  Memory-bound: ~2.15 GB of
// streaming traffic at 23.3 TB/s -> ~92 us floor; WMMA keeps the ~187 TF/s
// fp32 requirement off the critical path.
__global__ __launch_bounds__(256) void layer_wmma_f32(
    const float* __restrict__ x,
    const float* __restrict__ W,
    float* __restrict__ out,
    int nTiles)
{
    const int lane = (int)(threadIdx.x & 31u);
    const int m    = lane & 15;   // row-within-tile (A) / column (B,D)
    const int half = lane >> 4;   // K-pair selector for A/B fragments

    const int wavesPerBlock = (int)(blockDim.x >> 5);
    // Wave index is wave-uniform: pin it to an SGPR so loop control is scalar.
    const int wave = __builtin_amdgcn_readfirstlane(
        (int)blockIdx.x * wavesPerBlock + (int)(threadIdx.x >> 5));
    const int wstride = (int)gridDim.x * wavesPerBlock;

    // --- B fragments: W^T tiles for output columns [0,16) and [16,32). ---
    // f32 B-matrix 4x16 layout (mirror of A): lane l, vgpr j holds
    // B[2*(l>>4)+j][l&15] = W[o0 + (l&15)][kc*4 + 2*(l>>4) + j] -> contiguous float2.
    v2f bfrag[2][8];
#pragma unroll
    for (int h = 0; h < 2; ++h) {
        const float* wp = W + (size_t)((h << 4) + m) * 32 + (half << 1);
#pragma unroll
        for (int kc = 0; kc < 8; ++kc)
            bfrag[h][kc] = *(const v2f*)(wp + (kc << 2));
    }

    // Per-lane element offsets kept 32-bit off scalar bases (max index
    // N*32 = 2.7e8 << 2^31), stepped by a uniform stride each iteration.
    unsigned xoff = ((unsigned)wave * 16u + (unsigned)m) * 32u + (unsigned)(half << 1);
    unsigned ooff = ((unsigned)wave * 16u + (unsigned)(half << 3)) * 32u + (unsigned)m;
    const unsigned xstep = (unsigned)wstride * 16u * 32u;

    for (int t = wave; t < nTiles; t += wstride) {  // scalar loop, EXEC stays full
        const float* xp = x + xoff;

        // Prefetch next grid-stride tile's x into WGP-level cache (speculative,
        // OOB translation failures are silently dropped).
        __builtin_prefetch((const void*)(xp + (size_t)xstep), 0, 3);

        // --- A fragments: x tile 16x4 per chunk.  Lane l, vgpr j holds
        // A[l&15][2*(l>>4)+j] = x[n0 + (l&15)][kc*4 + 2*(l>>4) + j] -> float2. ---
        v2f a[8];
#pragma unroll
        for (int kc = 0; kc < 8; ++kc)
            a[kc] = *(const v2f*)(xp + (kc << 2));

        v8f acc0 = {};
        v8f acc1 = {};
#pragma unroll
        for (int kc = 0; kc < 8; ++kc) {
            acc0 = __builtin_amdgcn_wmma_f32_16x16x4_f32(
                false, a[kc], false, bfrag[0][kc], (short)0, acc0, false, false);
            acc1 = __builtin_amdgcn_wmma_f32_16x16x4_f32(
                false, a[kc], false, bfrag[1][kc], (short)0, acc1, false, false);
        }

        // --- ReLU + streaming store.  D layout: vgpr r, lane l ->
        // out[n0 + r + 8*(l>>4)][h*16 + (l&15)].  Half-wave rows are 64B
        // contiguous; select form folds to a single v_max_num_f32 per value. ---
        float* op = out + ooff;
#pragma unroll
        for (int r = 0; r < 8; ++r) {
            float v0 = acc0[r];
            float v1 = acc1[r];
            v0 = v0 > 0.0f ? v0 : 0.0f;
            v1 = v1 > 0.0f ? v1 : 0.0f;
            __builtin_nontemporal_store(v0, op + (unsigned)(r * 32));
            __builtin_nontemporal_store(v1, op + (unsigned)(r * 32 + 16));
        }

        xoff += xstep;
        ooff += xstep;
    }
}

extern "C" void kernel_launch(void* const* d_in, const int* in_sizes, int n_in,
                              void* d_out, int out_size, void* d_ws, size_t ws_size,
                              hipStream_t stream) {
    const float* x   = (const float*)d_in[0];  // [N, 32] fp32
    const float* W   = (const float*)d_in[1];  // [32, 32] fp32
    float*       out = (float*)d_out;          // [N, 32] fp32

    const int rows   = in_sizes[0] / 32;       // N
    const int nTiles = rows >> 4;              // N/16 (N is a multiple of 16)

    const int block = 256;                     // 8 waves per workgroup
    // ~8 tiles per wave so B fragments (W) amortize across the grid-stride loop.
    long waves_needed = ((long)nTiles + 7) / 8;
    long grid = (waves_needed + 7) / 8;        // 8 waves per block
    if (grid < 1) grid = 1;
    if (grid > 1048576) grid = 1048576;

    layer_wmma_f32<<<(int)grid, block, 0, stream>>>(x, W, out, nTiles);
}